// WeightedHausdorffDistance_57793079935359
// MI455X (gfx1250) — compile-verified
//
#include <hip/hip_runtime.h>

// ---- problem constants (match reference) ----
#define HH 384
#define WW 384
#define BB 4
#define MM 192
#define NN (HH * WW)                    // 147456 pixels
#define EPSF 1e-6f
#define MAX_DIST 543.058001f            // sqrt(384^2 + 384^2)
#define EPS_OVER_MAXD (EPSF / MAX_DIST)

typedef __attribute__((ext_vector_type(2))) float v2f;
typedef __attribute__((ext_vector_type(8))) float v8f;

#define WAVES_PER_BLOCK 8
#define ROWS_PER_BLOCK (WAVES_PER_BLOCK * 16)     // 128 pixel rows per block
#define BLOCKS_PER_IMG (NN / ROWS_PER_BLOCK)      // 1152

// ---------------- init scratch ----------------
__global__ void whd_init(float* __restrict__ t1num, float* __restrict__ psum,
                         unsigned int* __restrict__ colmin) {
  int t = threadIdx.x;
  if (t < BB) { t1num[t] = 0.f; psum[t] = 0.f; }
  for (int i = t; i < BB * MM; i += blockDim.x) colmin[i] = 0x7F800000u; // +inf
}

// ---------------- main: WMMA distance tiles + reductions ----------------
__global__ __launch_bounds__(256) void whd_main(const float* __restrict__ prob,
                                                const float* __restrict__ gt,
                                                const float* __restrict__ osz,
                                                float* __restrict__ t1num,
                                                float* __restrict__ psum,
                                                unsigned int* __restrict__ gcolmin) {
  // B-fragment table, pre-split by lane half so the inner loop is branch-free:
  //   sB[c]        = (-2*gy[c], -2*gx[c])   for lanes 0-15  (K rows 0,1)
  //   sB[192 + c]  = ( 1.0,     gy^2+gx^2 ) for lanes 16-31 (K rows 2,3)
  __shared__ float2 sB[2 * MM];
  __shared__ unsigned int scm[MM];

  const int b   = blockIdx.y;
  const int tid = threadIdx.x;
  const float normY = osz[2 * b + 0] * (1.0f / (float)HH);
  const float normX = osz[2 * b + 1] * (1.0f / (float)WW);

  if (tid < MM) {
    float gy = gt[(b * MM + tid) * 2 + 0] * normY;
    float gx = gt[(b * MM + tid) * 2 + 1] * normX;
    sB[tid]      = make_float2(-2.0f * gy, -2.0f * gx);
    sB[MM + tid] = make_float2(1.0f, gy * gy + gx * gx);
    scm[tid] = 0x7F800000u;
  }
  __syncthreads();

  const int wave = tid >> 5;
  const int lane = tid & 31;
  const int grp  = lane >> 4;        // which 16-lane half
  const int lrow = lane & 15;
  const int baseN = (blockIdx.x * WAVES_PER_BLOCK + wave) * 16;

  // ---- A fragment: A[r] = [Y, X, Y^2+X^2, 1]  (16x4 f32 layout:
  //      lanes 0-15 hold K=0,1 in v0,v1; lanes 16-31 hold K=2,3) ----
  const int n  = baseN + lrow;
  const float fy  = (float)(n / WW) * normY;
  const float fx  = (float)(n % WW) * normX;
  const float asq = fy * fy + fx * fx;
  v2f afrag;
  afrag.x = grp ? asq  : fy;
  afrag.y = grp ? 1.0f : fx;

  // p for this strip's 16 rows: lanes r and r+16 both hold p[baseN+r].
  const float pv = prob[b * NN + n];
  float pr[8], invden[8], rowAcc[8];
#pragma unroll
  for (int i = 0; i < 8; ++i) {
    float p = __shfl(pv, i + 8 * grp, 32);       // p of row handled by VGPR slot i
    pr[i] = p;
    float p2 = p * p;
    invden[i] = 1.0f / (p2 * p2 + EPS_OVER_MAXD);
    rowAcc[i] = 3.4e38f;
  }

  const int bidx = grp * MM + lrow;              // per-lane base into sB

  // ---- sweep 12 column tiles of 16 GT points each ----
#pragma unroll 2
  for (int t = 0; t < MM / 16; ++t) {
    const int c = t * 16 + lrow;
    float2 bv = sB[bidx + t * 16];               // one unconditional ds_load_b64
    v2f bfrag; bfrag.x = bv.x; bfrag.y = bv.y;

    v8f dacc = {};
    // d2[r][c] tile: one matrix op per 16x16 tile
    dacc = __builtin_amdgcn_wmma_f32_16x16x4_f32(false, afrag, false, bfrag,
                                                 (short)0, dacc, false, false);
    float cmin = 3.4e38f;
#pragma unroll
    for (int i = 0; i < 8; ++i) {
      // d2 < 0 only via rounding cancellation (|d2| ~ 0), so sqrt(|d2|) is
      // equivalent — and |src| is a free VALU input modifier on v_sqrt_f32.
      float d = __builtin_amdgcn_sqrtf(__builtin_fabsf(dacc[i]));
      rowAcc[i] = fminf(rowAcc[i], d);                       // term1 row-min
      cmin = fminf(cmin, (d + EPSF) * invden[i]);            // term2 candidate
    }
    // Lanes l and l+16 hold the two 8-row halves of column c: let both issue
    // the LDS atomic min (same slot) — avoids a bpermute + exec diamond.
    atomicMin(&scm[c], __float_as_uint(cmin));
  }

  // ---- row-min across the 16 lanes of each half (covers all 192 columns) ----
#pragma unroll
  for (int off = 1; off < 16; off <<= 1) {
#pragma unroll
    for (int i = 0; i < 8; ++i)
      rowAcc[i] = fminf(rowAcc[i], __shfl_xor(rowAcc[i], off, 32));
  }
  // Every lane of half g now holds min-dist for rows 8g..8g+7.
  float localT1 = 0.f;
#pragma unroll
  for (int i = 0; i < 8; ++i) localT1 += pr[i] * rowAcc[i];

  float t1part = (lrow == 0) ? localT1 : 0.f;   // one contributor per half
  float ppart  = (grp == 0) ? pv : 0.f;         // each row's p counted once
#pragma unroll
  for (int off = 16; off >= 1; off >>= 1) {
    t1part += __shfl_xor(t1part, off, 32);
    ppart  += __shfl_xor(ppart,  off, 32);
  }
  if (lane == 0) {
    atomicAdd(&t1num[b], t1part);
    atomicAdd(&psum[b],  ppart);
  }

  __syncthreads();
  if (tid < MM) atomicMin(&gcolmin[b * MM + tid], scm[tid]);
}

// ---------------- finalize ----------------
__global__ void whd_finalize(const float* __restrict__ t1num, const float* __restrict__ psum,
                             const unsigned int* __restrict__ colmin, float* __restrict__ out) {
  __shared__ float red[128];
  float acc = 0.f;
  for (int idx = threadIdx.x; idx < BB * MM; idx += 128) {
    float v = __uint_as_float(colmin[idx]);
    v = fminf(fmaxf(v, 0.f), MAX_DIST);
    acc += v;
  }
  red[threadIdx.x] = acc;
  __syncthreads();
  for (int s = 64; s > 0; s >>= 1) {
    if ((int)threadIdx.x < s) red[threadIdx.x] += red[threadIdx.x + s];
    __syncthreads();
  }
  if (threadIdx.x == 0) {
    float t2 = red[0] / (float)(BB * MM);
    float t1 = 0.f;
    for (int b = 0; b < BB; ++b) t1 += t1num[b] / (psum[b] + EPSF);
    t1 *= (1.0f / (float)BB);
    out[0] = t1 + t2;
  }
}

// ---------------- host launcher ----------------
extern "C" void kernel_launch(void* const* d_in, const int* in_sizes, int n_in,
                              void* d_out, int out_size, void* d_ws, size_t ws_size,
                              hipStream_t stream) {
  const float* prob = (const float*)d_in[0];   // [B,H,W]
  const float* gt   = (const float*)d_in[1];   // [B,M,2]
  const float* osz  = (const float*)d_in[2];   // [B,2]

  float* t1num = (float*)d_ws;                         // 8 floats (4 used, padded)
  float* psum  = t1num + 8;                            // 8 floats
  unsigned int* colmin = (unsigned int*)(psum + 8);    // B*M uints

  whd_init<<<1, 256, 0, stream>>>(t1num, psum, colmin);

  dim3 grid(BLOCKS_PER_IMG, BB);
  whd_main<<<grid, 256, 0, stream>>>(prob, gt, osz, t1num, psum, colmin);

  whd_finalize<<<1, 128, 0, stream>>>(t1num, psum, colmin, (float*)d_out);
}